// RelGraphConvLayer_40278203301916
// MI455X (gfx1250) — compile-verified
//
#include <hip/hip_runtime.h>
#include <hip/hip_bf16.h>

typedef __attribute__((ext_vector_type(2))) float v2f;
typedef __attribute__((ext_vector_type(8))) float v8f;

#define NNODES_FEAT 128   // DIN == DOUT == 128

// ---------------------------------------------------------------------------
// Zero a float buffer (float4 granularity; count must be divisible by 4).
// ---------------------------------------------------------------------------
__global__ void zero_kernel(float* __restrict__ p, int count4) {
    int i = blockIdx.x * blockDim.x + threadIdx.x;
    if (i < count4) {
        ((float4*)p)[i] = make_float4(0.f, 0.f, 0.f, 0.f);
    }
}

// ---------------------------------------------------------------------------
// Edge scatter: thread t handles edge (t>>5) and 4-float chunk (t&31).
// agg[dst] += x[src] (f32 atomics, L2-resident), chunk 0 also bumps deg[dst].
// ---------------------------------------------------------------------------
__global__ void edge_scatter_kernel(const float* __restrict__ x,
                                    const int* __restrict__ src,
                                    const int* __restrict__ dst,
                                    float* __restrict__ agg,
                                    float* __restrict__ deg,
                                    int E) {
    int t = blockIdx.x * blockDim.x + threadIdx.x;
    int e = t >> 5;
    if (e >= E) return;
    int c = (t & 31) * 4;
    int s = src[e];
    int d = dst[e];
    float4 v = *(const float4*)(x + (size_t)s * NNODES_FEAT + c);
    float* p = agg + (size_t)d * NNODES_FEAT + c;
    atomicAdd(p + 0, v.x);
    atomicAdd(p + 1, v.y);
    atomicAdd(p + 2, v.z);
    atomicAdd(p + 3, v.w);
    if (c == 0) atomicAdd(deg + d, 1.0f);
}

// ---------------------------------------------------------------------------
// WMMA GEMM: out[rows, 128] (+)= (A[rows,128] * inv_deg) @ W[128,128] (+ bias)
// Block = 256 threads = 8 waves; wave w owns rows [blk*128 + w*16, +16).
// W staged in LDS. Uses V_WMMA_F32_16X16X4_F32, K stepped in chunks of 4.
//
// Per-lane fragment layout (ISA 7.12.2, f32 16x16x4):
//   A (16x4): lane l -> M = l&15, half h = l>>4; vgpr v holds K = 4k + 2h + v
//   B (4x16): lane l -> N = l&15; vgpr v holds K = 4k + 2h + v
//   C/D (16x16): vgpr j -> M = j + 8h, N = l&15
// ---------------------------------------------------------------------------
__global__ __launch_bounds__(256) void gemm_wmma_kernel(
    const float* __restrict__ A,      // [N,128]  (agg or x)
    const float* __restrict__ W,      // [128,128] row-major (K,N)
    const float* __restrict__ deg,    // [N] or nullptr (self-loop pass)
    const float* __restrict__ bias,   // [128] or nullptr
    float* __restrict__ out,          // [N,128]
    int N, int accumulate) {

    __shared__ float sW[128 * 128];   // 64 KB

    // Cooperative load of W into LDS (float4, 16 per thread).
    for (int i = threadIdx.x; i < 128 * 128 / 4; i += 256) {
        ((float4*)sW)[i] = ((const float4*)W)[i];
    }
    __syncthreads();

    const int wave = threadIdx.x >> 5;
    const int lane = threadIdx.x & 31;
    const int m16  = lane & 15;
    const int hi   = lane >> 4;

    const int row0 = blockIdx.x * 128 + wave * 16;
    int arow_idx = row0 + m16;
    if (arow_idx > N - 1) arow_idx = N - 1;    // clamp loads; stores guarded

    float inv = 1.0f;
    if (deg) {
        float dg = deg[arow_idx];
        inv = 1.0f / fmaxf(dg, 1.0f);
    }

    // Load & normalize this lane's A fragments: 32 K-steps of float2.
    v2f a[32];
    const float* arow = A + (size_t)arow_idx * 128;
    #pragma unroll
    for (int k = 0; k < 32; ++k) {
        int kk = k * 4 + 2 * hi;
        a[k].x = arow[kk + 0] * inv;
        a[k].y = arow[kk + 1] * inv;
    }

    // 8 output N-tiles of 16 columns each.
    for (int nt = 0; nt < 8; ++nt) {
        const int n0 = nt * 16;
        v8f c = {};
        #pragma unroll
        for (int k = 0; k < 32; ++k) {
            int kk = k * 4 + 2 * hi;
            v2f b;
            b.x = sW[(kk + 0) * 128 + n0 + m16];
            b.y = sW[(kk + 1) * 128 + n0 + m16];
            c = __builtin_amdgcn_wmma_f32_16x16x4_f32(
                    /*neg_a=*/false, a[k], /*neg_b=*/false, b,
                    /*c_mod=*/(short)0, c, /*reuse_a=*/false, /*reuse_b=*/false);
        }
        #pragma unroll
        for (int j = 0; j < 8; ++j) {
            int r = row0 + j + 8 * hi;
            if (r < N) {
                float* p = out + (size_t)r * 128 + n0 + m16;
                float v = c[j];
                if (accumulate) v += *p;
                else if (bias)  v += bias[n0 + m16];
                *p = v;
            }
        }
    }
}

// ---------------------------------------------------------------------------
// In-place ReLU (float4 granularity).
// ---------------------------------------------------------------------------
__global__ void relu_kernel(float* __restrict__ p, int count4) {
    int i = blockIdx.x * blockDim.x + threadIdx.x;
    if (i < count4) {
        float4 v = ((float4*)p)[i];
        v.x = fmaxf(v.x, 0.f);
        v.y = fmaxf(v.y, 0.f);
        v.z = fmaxf(v.z, 0.f);
        v.w = fmaxf(v.w, 0.f);
        ((float4*)p)[i] = v;
    }
}

// ---------------------------------------------------------------------------
// Inputs (setup_inputs order):
//  0: x [N,128] f32
//  1: src_r0  2: dst_r0  3: src_r1  4: dst_r1  5: src_r2  6: dst_r2  (int32, E)
//  7: W_r0  8: W_r1  9: W_r2  10: W_loop  [128,128] f32
//  11: b_loop [128] f32
// Output: [N,128] f32
// ---------------------------------------------------------------------------
extern "C" void kernel_launch(void* const* d_in, const int* in_sizes, int n_in,
                              void* d_out, int out_size, void* d_ws, size_t ws_size,
                              hipStream_t stream) {
    const float* x      = (const float*)d_in[0];
    const int*   src[3] = {(const int*)d_in[1], (const int*)d_in[3], (const int*)d_in[5]};
    const int*   dst[3] = {(const int*)d_in[2], (const int*)d_in[4], (const int*)d_in[6]};
    const float* Wr[3]  = {(const float*)d_in[7], (const float*)d_in[8], (const float*)d_in[9]};
    const float* W_loop = (const float*)d_in[10];
    const float* b_loop = (const float*)d_in[11];
    float*       out    = (float*)d_out;

    const int N = in_sizes[0] / NNODES_FEAT;   // 50000
    const int E = in_sizes[1];                 // 500000

    float* agg = (float*)d_ws;                 // [N,128]
    float* deg = agg + (size_t)N * NNODES_FEAT;  // [N]

    const int gemm_blocks = (N + 127) / 128;

    // 1) out = x @ W_loop + b_loop   (init pass, no normalization)
    gemm_wmma_kernel<<<gemm_blocks, 256, 0, stream>>>(
        x, W_loop, /*deg=*/nullptr, /*bias=*/b_loop, out, N, /*accumulate=*/0);

    // 2) Per relation: zero agg/deg, scatter, accumulate GEMM into out.
    const int zero4   = (N * NNODES_FEAT + N) / 4;
    const int zblocks = (zero4 + 255) / 256;
    const int eblocks = ((E * 32) + 255) / 256;
    for (int r = 0; r < 3; ++r) {
        zero_kernel<<<zblocks, 256, 0, stream>>>(agg, zero4);
        edge_scatter_kernel<<<eblocks, 256, 0, stream>>>(x, src[r], dst[r], agg, deg, E);
        gemm_wmma_kernel<<<gemm_blocks, 256, 0, stream>>>(
            agg, Wr[r], deg, /*bias=*/nullptr, out, N, /*accumulate=*/1);
    }

    // 3) ReLU in place.
    const int relu4   = (N * NNODES_FEAT) / 4;
    const int rblocks = (relu4 + 255) / 256;
    relu_kernel<<<rblocks, 256, 0, stream>>>(out, relu4);
}